// MSCLoss_8675833938569
// MI455X (gfx1250) — compile-verified
//
#include <hip/hip_runtime.h>
#include <math.h>

#define NN 4096
#define DD 256
#define KTOP 10
#define MTOP 5
#define MUV 2048
#define NCLS 31

typedef __attribute__((ext_vector_type(2))) float v2f;
typedef __attribute__((ext_vector_type(8))) float v8f;

// ---------------- Phase 1a: row squared norms ----------------
__global__ void __launch_bounds__(256)
sqnorm_kernel(const float* __restrict__ src, const float* __restrict__ tgt,
              float* __restrict__ sqs, float* __restrict__ sqt) {
    __shared__ float red[256];
    const int r = blockIdx.x, tid = threadIdx.x;
    float a = src[r * DD + tid];
    red[tid] = a * a;
    __syncthreads();
    for (int s = 128; s > 0; s >>= 1) { if (tid < s) red[tid] += red[tid + s]; __syncthreads(); }
    if (tid == 0) sqs[r] = red[0];
    __syncthreads();
    float b = tgt[r * DD + tid];
    red[tid] = b * b;
    __syncthreads();
    for (int s = 128; s > 0; s >>= 1) { if (tid < s) red[tid] += red[tid + s]; __syncthreads(); }
    if (tid == 0) sqt[r] = red[0];
}

// ---------------- Phase 1b: WMMA GEMM + similarity epilogue ----------------
// simT[t][s] = 1 / (sqrt(max(|t|^2 + |s|^2 - 2 <tgt_t, src_s>, 0)) + 1)
// One wave per 32x32 tile; 2x2 grid of 16x16 fragments; K-step 4 (fp32 WMMA).
__global__ void __launch_bounds__(32)
gemm_sim_kernel(const float* __restrict__ src, const float* __restrict__ tgt,
                const float* __restrict__ sqs, const float* __restrict__ sqt,
                float* __restrict__ simT) {
    const int lane  = threadIdx.x;           // 0..31, full wave
    const int mrow  = lane & 15;             // row within 16 (A) / col within 16 (B)
    const int khalf = (lane >> 4) << 1;      // lanes 16-31 hold K=2,3
    const int s0 = blockIdx.x * 32;
    const int t0 = blockIdx.y * 32;

    v8f c00 = {}, c01 = {}, c10 = {}, c11 = {};
    const float* a0p = tgt + (size_t)(t0 + mrow) * DD + khalf;
    const float* a1p = tgt + (size_t)(t0 + 16 + mrow) * DD + khalf;
    const float* b0p = src + (size_t)(s0 + mrow) * DD + khalf;
    const float* b1p = src + (size_t)(s0 + 16 + mrow) * DD + khalf;

    #pragma unroll 4
    for (int k = 0; k < DD; k += 4) {
        v2f a0 = *(const v2f*)(a0p + k);
        v2f a1 = *(const v2f*)(a1p + k);
        v2f b0 = *(const v2f*)(b0p + k);
        v2f b1 = *(const v2f*)(b1p + k);
        c00 = __builtin_amdgcn_wmma_f32_16x16x4_f32(false, a0, false, b0, (short)0, c00, false, false);
        c01 = __builtin_amdgcn_wmma_f32_16x16x4_f32(false, a0, false, b1, (short)0, c01, false, false);
        c10 = __builtin_amdgcn_wmma_f32_16x16x4_f32(false, a1, false, b0, (short)0, c10, false, false);
        c11 = __builtin_amdgcn_wmma_f32_16x16x4_f32(false, a1, false, b1, (short)0, c11, false, false);
    }

    // C/D layout: VGPR r, lane L: M = r + (L<16 ? 0 : 8), N = L%16
    const int ncol  = lane & 15;
    const int mbase = (lane >> 4) * 8;
    const int ss0 = s0 + ncol;
    const int ss1 = s0 + 16 + ncol;
    const float q_s0 = sqs[ss0];
    const float q_s1 = sqs[ss1];
    #pragma unroll
    for (int r = 0; r < 8; ++r) {
        const int tA = t0 + mbase + r;
        const int tB = t0 + 16 + mbase + r;
        const float qtA = sqt[tA];
        const float qtB = sqt[tB];
        float d;
        d = fmaxf(qtA + q_s0 - 2.0f * c00[r], 0.0f);
        simT[(size_t)tA * NN + ss0] = 1.0f / (sqrtf(d) + 1.0f);
        d = fmaxf(qtA + q_s1 - 2.0f * c01[r], 0.0f);
        simT[(size_t)tA * NN + ss1] = 1.0f / (sqrtf(d) + 1.0f);
        d = fmaxf(qtB + q_s0 - 2.0f * c10[r], 0.0f);
        simT[(size_t)tB * NN + ss0] = 1.0f / (sqrtf(d) + 1.0f);
        d = fmaxf(qtB + q_s1 - 2.0f * c11[r], 0.0f);
        simT[(size_t)tB * NN + ss1] = 1.0f / (sqrtf(d) + 1.0f);
    }
}

// ---------------- Phase 2: per-target kNN vote + top-5 same/diff sums ----------------
__global__ void __launch_bounds__(256)
target_stats_kernel(const float* __restrict__ simT, const int* __restrict__ labels,
                    int* __restrict__ assigned, float* __restrict__ score) {
    __shared__ float s_sim[NN];
    __shared__ int   s_lab[NN];
    __shared__ float s_rv[256];
    __shared__ int   s_ri[256];
    __shared__ int   s_counts[NCLS];
    __shared__ int   s_pidx[KTOP];
    __shared__ float s_pval[KTOP];
    __shared__ int   s_assigned;

    const int t = blockIdx.x, tid = threadIdx.x;
    for (int i = 0; i < NN / 256; ++i) {
        const int idx = tid + i * 256;
        s_sim[idx] = simT[(size_t)t * NN + idx];
        s_lab[idx] = labels[idx];
    }
    if (tid < NCLS) s_counts[tid] = 0;
    __syncthreads();

    // top-10 by iterative max extraction (tie -> lowest index, like jax top_k)
    for (int r = 0; r < KTOP; ++r) {
        float bv = -2.0f; int bi = 0;
        for (int i = 0; i < NN / 256; ++i) {
            const int idx = tid + i * 256;
            const float v = s_sim[idx];
            if (v > bv || (v == bv && idx < bi)) { bv = v; bi = idx; }
        }
        s_rv[tid] = bv; s_ri[tid] = bi;
        __syncthreads();
        for (int s = 128; s > 0; s >>= 1) {
            if (tid < s) {
                const float ov = s_rv[tid + s]; const int oi = s_ri[tid + s];
                if (ov > s_rv[tid] || (ov == s_rv[tid] && oi < s_ri[tid])) { s_rv[tid] = ov; s_ri[tid] = oi; }
            }
            __syncthreads();
        }
        if (tid == 0) {
            const int idx = s_ri[0];
            s_pidx[r] = idx; s_pval[r] = s_rv[0];
            s_sim[idx] = -1.0f;            // consume
            s_counts[s_lab[idx]]++;
        }
        __syncthreads();
    }
    if (tid == 0) {
        for (int r = 0; r < KTOP; ++r) s_sim[s_pidx[r]] = s_pval[r];  // restore
        int best = 0, bc = s_counts[0];
        for (int c = 1; c < NCLS; ++c) if (s_counts[c] > bc) { bc = s_counts[c]; best = c; }
        s_assigned = best;
        assigned[t] = best;
    }
    __syncthreads();
    const int asg = s_assigned;

    // top-5 sums over same-class (phase 0) then diff-class (phase 1) sims
    float sums[2];
    for (int phase = 0; phase < 2; ++phase) {
        float acc = 0.0f;
        for (int r = 0; r < MTOP; ++r) {
            float bv = 0.0f; int bi = -1;
            for (int i = 0; i < NN / 256; ++i) {
                const int idx = tid + i * 256;
                const bool samec = (s_lab[idx] == asg);
                const bool want = (phase == 0) ? samec : !samec;
                const float v = s_sim[idx];   // consumed entries are -1 (< 0)
                if (want && v > 0.0f) {
                    if (v > bv || (v == bv && (bi < 0 || idx < bi))) { bv = v; bi = idx; }
                }
            }
            s_rv[tid] = bv; s_ri[tid] = bi;
            __syncthreads();
            for (int s = 128; s > 0; s >>= 1) {
                if (tid < s) {
                    const float ov = s_rv[tid + s]; const int oi = s_ri[tid + s];
                    const float cv = s_rv[tid];     const int ci = s_ri[tid];
                    const bool take = (ov > cv) || (ov == cv && oi >= 0 && (ci < 0 || oi < ci));
                    if (take) { s_rv[tid] = ov; s_ri[tid] = oi; }
                }
                __syncthreads();
            }
            if (tid == 0 && s_rv[0] > 0.0f) s_sim[s_ri[0]] = -1.0f;   // consume winner
            acc += (s_rv[0] > 0.0f) ? s_rv[0] : 0.0f;                 // all threads track acc
            __syncthreads();
        }
        sums[phase] = acc;
    }
    if (tid == 0) score[t] = sums[0] / sums[1];
}

// ---------------- Phase 3: top-MU targets via single-block bitonic sort ----------------
__global__ void __launch_bounds__(1024)
topmu_kernel(const float* __restrict__ score, const int* __restrict__ assigned,
             int* __restrict__ top_tgt, int* __restrict__ flab) {
    __shared__ float sv[NN];
    __shared__ int   si[NN];
    const int tid = threadIdx.x;
    for (int i = 0; i < NN / 1024; ++i) { const int idx = tid + i * 1024; sv[idx] = score[idx]; si[idx] = idx; }
    __syncthreads();
    for (int k = 2; k <= NN; k <<= 1) {
        for (int j = k >> 1; j > 0; j >>= 1) {
            for (int i = 0; i < NN / 1024; ++i) {
                const int idx = tid + i * 1024;
                const int ixj = idx ^ j;
                if (ixj > idx) {
                    const float va = sv[idx], vb = sv[ixj];
                    const int   ia = si[idx], ib = si[ixj];
                    const bool a_first = (va > vb) || (va == vb && ia < ib);  // descending, tie->low idx
                    const bool up = ((idx & k) == 0);
                    const bool do_swap = up ? !a_first : a_first;
                    if (do_swap) { sv[idx] = vb; sv[ixj] = va; si[idx] = ib; si[ixj] = ia; }
                }
            }
            __syncthreads();
        }
    }
    for (int i = 0; i < MUV / 1024; ++i) {
        const int j = tid + i * 1024;
        const int tt = si[j];
        top_tgt[j] = tt;
        flab[j] = assigned[tt];
    }
}

// ---------------- Phase 4: online softmax over selected targets ----------------
__global__ void __launch_bounds__(256)
loss_partial_kernel(const float* __restrict__ simT, const int* __restrict__ labels,
                    const int* __restrict__ top_tgt, const int* __restrict__ flab,
                    float* __restrict__ psum, float* __restrict__ pcnt) {
    __shared__ int s_tt[256], s_fl[256];
    __shared__ float red[256];
    const int tid = threadIdx.x;
    const int s = blockIdx.x * 256 + tid;
    const int lab = labels[s];
    float m = -3.402823e38f, den = 0.0f, num = 0.0f;
    int nsame = 0;
    for (int c = 0; c < MUV; c += 256) {
        s_tt[tid] = top_tgt[c + tid];
        s_fl[tid] = flab[c + tid];
        __syncthreads();
        for (int j = 0; j < 256; ++j) {
            const float x = simT[(size_t)s_tt[j] * NN + s];   // coalesced across threads
            const bool mk = (s_fl[j] == lab);
            nsame += mk ? 1 : 0;
            const float nm = fmaxf(m, x);
            const float scale = expf(m - nm);
            const float e = expf(x - nm);
            den = den * scale + e;
            num = num * scale + (mk ? e : 0.0f);
            m = nm;
        }
        __syncthreads();
    }
    const bool valid = (nsame > 0) && (nsame < MUV);
    const float ls = valid ? -logf(num / den) : 0.0f;
    const float lc = valid ? 1.0f : 0.0f;
    red[tid] = ls;
    __syncthreads();
    for (int r = 128; r > 0; r >>= 1) { if (tid < r) red[tid] += red[tid + r]; __syncthreads(); }
    if (tid == 0) psum[blockIdx.x] = red[0];
    __syncthreads();
    red[tid] = lc;
    __syncthreads();
    for (int r = 128; r > 0; r >>= 1) { if (tid < r) red[tid] += red[tid + r]; __syncthreads(); }
    if (tid == 0) pcnt[blockIdx.x] = red[0];
}

// ---------------- Phase 5: deterministic final reduction ----------------
__global__ void finalize_kernel(const float* __restrict__ psum, const float* __restrict__ pcnt,
                                float* __restrict__ out, int nb) {
    if (threadIdx.x == 0 && blockIdx.x == 0) {
        float s = 0.0f, c = 0.0f;
        for (int i = 0; i < nb; ++i) { s += psum[i]; c += pcnt[i]; }
        out[0] = s / c;
    }
}

extern "C" void kernel_launch(void* const* d_in, const int* in_sizes, int n_in,
                              void* d_out, int out_size, void* d_ws, size_t ws_size,
                              hipStream_t stream) {
    const float* src    = (const float*)d_in[0];  // src_features  (N, D) f32
    const int*   labels = (const int*)  d_in[1];  // src_labels    (N,)   i32
    const float* tgt    = (const float*)d_in[2];  // tgt_features  (N, D) f32
    float* out = (float*)d_out;

    char* ws = (char*)d_ws;
    float* simT     = (float*)ws;                                    // N*N floats (64 MB)
    float* sqs      = (float*)(ws + (size_t)NN * NN * sizeof(float));
    float* sqt      = sqs + NN;
    float* score    = sqt + NN;
    int*   assigned = (int*)(score + NN);
    int*   top_tgt  = assigned + NN;
    int*   flab     = top_tgt + MUV;
    float* psum     = (float*)(flab + MUV);
    float* pcnt     = psum + (NN / 256);

    sqnorm_kernel<<<NN, 256, 0, stream>>>(src, tgt, sqs, sqt);
    gemm_sim_kernel<<<dim3(NN / 32, NN / 32), 32, 0, stream>>>(src, tgt, sqs, sqt, simT);
    target_stats_kernel<<<NN, 256, 0, stream>>>(simT, labels, assigned, score);
    topmu_kernel<<<1, 1024, 0, stream>>>(score, assigned, top_tgt, flab);
    loss_partial_kernel<<<NN / 256, 256, 0, stream>>>(simT, labels, top_tgt, flab, psum, pcnt);
    finalize_kernel<<<1, 32, 0, stream>>>(psum, pcnt, out, NN / 256);
}